// GeneHomologyGNN_18743237280102
// MI455X (gfx1250) — compile-verified
//
#include <hip/hip_runtime.h>
#include <hip/hip_bf16.h>
#include <math.h>

typedef __attribute__((ext_vector_type(2))) float v2f;
typedef __attribute__((ext_vector_type(8))) float v8f;

#define EMB 128
#define HID 128

// ---------------------------------------------------------------------------
// feats[i] = [ emb[up_idx[i]] | emb[i] | emb[down_idx[i]] ],  emb[i] = table[gene_ids[i]]
// grid: N blocks of 128 threads
__global__ void gather_feats_kernel(const float* __restrict__ emb_table,
                                    const int* __restrict__ gene_ids,
                                    const int* __restrict__ up_idx,
                                    const int* __restrict__ down_idx,
                                    float* __restrict__ feats) {
    int i = blockIdx.x;
    int j = threadIdx.x;           // 0..127
    int g_self = gene_ids[i];
    int g_up   = gene_ids[up_idx[i]];
    int g_dn   = gene_ids[down_idx[i]];
    size_t base = (size_t)i * (3 * EMB);
    feats[base + j]            = emb_table[(size_t)g_up   * EMB + j];
    feats[base + EMB + j]      = emb_table[(size_t)g_self * EMB + j];
    feats[base + 2 * EMB + j]  = emb_table[(size_t)g_dn   * EMB + j];
}

// ---------------------------------------------------------------------------
__global__ void init_deg_kernel(float* __restrict__ deg, int n) {
    int i = blockIdx.x * blockDim.x + threadIdx.x;
    if (i < n) deg[i] = 1.0f;      // self-loop contribution
}

__global__ void accum_deg_kernel(const int* __restrict__ col, float* __restrict__ deg, int E) {
    int e = blockIdx.x * blockDim.x + threadIdx.x;
    if (e < E) atomicAdd(&deg[col[e]], 1.0f);
}

__global__ void rsqrt_inplace_kernel(float* __restrict__ deg, int n) {
    int i = blockIdx.x * blockDim.x + threadIdx.x;
    if (i < n) deg[i] = rsqrtf(deg[i]);
}

// ---------------------------------------------------------------------------
// C[M x 128] = A[M x K] * B[K x 128], full fp32 via V_WMMA_F32_16X16X4_F32.
// One wave -> one 16x16 tile. Block = (32,8): 8 waves cover all 8 col tiles of
// one 16-row tile. grid.x = M/16 (M is a multiple of 16). EXEC all-ones.
//
// A operand layout (16x4 f32): lanes 0-15 hold {K=k, K=k+1} for row M=lane,
// lanes 16-31 hold {K=k+2, K=k+3}. B (4x16) mirrored across lanes by N.
// C/D: VGPR j -> M=j (lanes 0-15) / M=j+8 (lanes 16-31), N = lane%16.
__global__ void gemm_wmma_f32_kernel(const float* __restrict__ A,
                                     const float* __restrict__ B,
                                     float* __restrict__ C, int K) {
    const int NC   = HID;
    int lane = threadIdx.x;        // 0..31
    int half = lane >> 4;          // 0 or 1
    int lm   = lane & 15;
    int row0 = blockIdx.x * 16;
    int col0 = threadIdx.y * 16;   // wave id 0..7

    v8f acc = {};
    const float* arow = A + (size_t)(row0 + lm) * K;
    for (int k = 0; k < K; k += 4) {
        int ka = k + 2 * half;
        v2f a;
        a.x = arow[ka];
        a.y = arow[ka + 1];
        v2f b;
        b.x = B[(size_t)ka * NC + col0 + lm];
        b.y = B[(size_t)(ka + 1) * NC + col0 + lm];
        acc = __builtin_amdgcn_wmma_f32_16x16x4_f32(
            /*neg_a=*/false, a, /*neg_b=*/false, b,
            /*c_mod=*/(short)0, acc, /*reuse_a=*/false, /*reuse_b=*/false);
    }
#pragma unroll
    for (int j = 0; j < 8; ++j) {
        C[(size_t)(row0 + j + 8 * half) * NC + col0 + lm] = acc[j];
    }
}

// ---------------------------------------------------------------------------
// out[i][j] = bias[j]   (pre-fill so scatter produces scatter_sum + b)
__global__ void fill_bias_kernel(float* __restrict__ out, const float* __restrict__ bias) {
    out[(size_t)blockIdx.x * HID + threadIdx.x] = bias[threadIdx.x];
}

// One wave per (edge or self-loop). lane handles 4 of 128 features.
// block = (32,8) -> 8 edges per block.
__global__ void scatter_edges_kernel(const float* __restrict__ h,
                                     const int* __restrict__ row,
                                     const int* __restrict__ col,
                                     const float* __restrict__ dinv,
                                     float* __restrict__ out, int E, int n) {
    int e = blockIdx.x * 8 + threadIdx.y;
    if (e >= E + n) return;
    int r, c;
    if (e < E) { r = row[e]; c = col[e]; }
    else       { r = c = e - E; }
    float coef = dinv[r] * dinv[c];
    int j = threadIdx.x * 4;
    const float4 hv = *(const float4*)(h + (size_t)r * HID + j);
    float* o = out + (size_t)c * HID + j;
    atomicAdd(o + 0, hv.x * coef);
    atomicAdd(o + 1, hv.y * coef);
    atomicAdd(o + 2, hv.z * coef);
    atomicAdd(o + 3, hv.w * coef);
}

__global__ void relu_inplace_kernel(float* __restrict__ x, size_t n) {
    size_t i = (size_t)blockIdx.x * blockDim.x + threadIdx.x;
    if (i < n) x[i] = fmaxf(x[i], 0.0f);
}

// ---------------------------------------------------------------------------
__global__ void zero128_kernel(float* __restrict__ p) { p[threadIdx.x] = 0.0f; }

// block: 128 threads (one per column), each block sums 256 rows, one atomic per col.
__global__ void pool_reduce_kernel(const float* __restrict__ x, float* __restrict__ pooled, int n) {
    int j = threadIdx.x;
    int i0 = blockIdx.x * 256;
    int i1 = min(i0 + 256, n);
    float s = 0.0f;
    for (int i = i0; i < i1; ++i) s += x[(size_t)i * HID + j];
    atomicAdd(&pooled[j], s);
}

// single block, 128 threads: sigmoid(relu(pooled@Wc1 + bc1) @ Wc2 + bc2)
__global__ void head_kernel(const float* __restrict__ pooled,
                            const float* __restrict__ Wc1, const float* __restrict__ bc1,
                            const float* __restrict__ Wc2, const float* __restrict__ bc2,
                            float* __restrict__ out, int n) {
    __shared__ float p[HID];
    __shared__ float hbuf[HID];
    int t = threadIdx.x;
    p[t] = pooled[t] / (float)n;       // mean pool
    __syncthreads();
    float acc = bc1[t];
    for (int k = 0; k < HID; ++k) acc += p[k] * Wc1[(size_t)k * HID + t];
    hbuf[t] = fmaxf(acc, 0.0f);
    __syncthreads();
    if (t == 0) {
        float s = bc2[0];
        for (int k = 0; k < HID; ++k) s += hbuf[k] * Wc2[k];
        out[0] = 1.0f / (1.0f + expf(-s));
    }
}

// ---------------------------------------------------------------------------
extern "C" void kernel_launch(void* const* d_in, const int* in_sizes, int n_in,
                              void* d_out, int out_size, void* d_ws, size_t ws_size,
                              hipStream_t stream) {
    const int*   gene_ids   = (const int*)d_in[0];
    const int*   edge_index = (const int*)d_in[1];   // [2, E]
    // d_in[2] edge_attr: unused; d_in[3] batch: unused (single graph)
    const int*   up_idx     = (const int*)d_in[4];
    const int*   down_idx   = (const int*)d_in[5];
    const float* emb_table  = (const float*)d_in[6];
    const float* W1         = (const float*)d_in[7];
    const float* b1         = (const float*)d_in[8];
    const float* W2         = (const float*)d_in[9];
    const float* b2         = (const float*)d_in[10];
    const float* Wc1        = (const float*)d_in[11];
    const float* bc1        = (const float*)d_in[12];
    const float* Wc2        = (const float*)d_in[13];
    const float* bc2        = (const float*)d_in[14];

    const int N = in_sizes[0];           // 50000 (multiple of 16)
    const int E = in_sizes[1] / 2;       // 800000
    const int* row = edge_index;
    const int* col = edge_index + E;

    // workspace carve (floats): feats | bufA | bufB | deg/dinv | pooled
    float* ws     = (float*)d_ws;
    float* feats  = ws;                                   // N * 384
    float* bufA   = feats + (size_t)N * 384;              // N * 128
    float* bufB   = bufA  + (size_t)N * HID;              // N * 128
    float* deg    = bufB  + (size_t)N * HID;              // N
    float* pooled = deg + N;                              // 128

    const int mtiles = N / 16;
    dim3 wblk(32, 8);

    // 1) node features [up | self | down]
    gather_feats_kernel<<<N, 128, 0, stream>>>(emb_table, gene_ids, up_idx, down_idx, feats);

    // 2) deg^-1/2 (in-degree of col + self loop)
    init_deg_kernel<<<(N + 255) / 256, 256, 0, stream>>>(deg, N);
    accum_deg_kernel<<<(E + 255) / 256, 256, 0, stream>>>(col, deg, E);
    rsqrt_inplace_kernel<<<(N + 255) / 256, 256, 0, stream>>>(deg, N);

    // 3) conv1: h = feats @ W1 ; x = relu(scatter(h*coef) + b1)
    gemm_wmma_f32_kernel<<<mtiles, wblk, 0, stream>>>(feats, W1, bufA, 3 * EMB);
    fill_bias_kernel<<<N, 128, 0, stream>>>(bufB, b1);
    scatter_edges_kernel<<<(E + N + 7) / 8, wblk, 0, stream>>>(bufA, row, col, deg, bufB, E, N);
    relu_inplace_kernel<<<(int)(((size_t)N * HID + 255) / 256), 256, 0, stream>>>(bufB, (size_t)N * HID);

    // 4) conv2: h = x @ W2 ; x2 = scatter(h*coef) + b2
    gemm_wmma_f32_kernel<<<mtiles, wblk, 0, stream>>>(bufB, W2, bufA, HID);
    fill_bias_kernel<<<N, 128, 0, stream>>>(bufB, b2);
    scatter_edges_kernel<<<(E + N + 7) / 8, wblk, 0, stream>>>(bufA, row, col, deg, bufB, E, N);

    // 5) mean pool + MLP head
    zero128_kernel<<<1, 128, 0, stream>>>(pooled);
    pool_reduce_kernel<<<(N + 255) / 256, 128, 0, stream>>>(bufB, pooled, N);
    head_kernel<<<1, 128, 0, stream>>>(pooled, Wc1, bc1, Wc2, bc2, (float*)d_out, N);
}